// VNL_Loss_ori_86285892977290
// MI455X (gfx1250) — compile-verified
//
#include <hip/hip_runtime.h>
#include <hip/hip_bf16.h>

#define Bc 8
#define Hc 768
#define Wc 1024
#define DELTA_COS 0.867f
#define DELTA_DIFF 0.005f
#define DELTA_Zf 0.0001f
#define INVALID_BITS 0xFFFFFFFFu

typedef __attribute__((ext_vector_type(2))) float v2f;
typedef __attribute__((ext_vector_type(8))) float v8f;

// Wave32 sum reduction on the matrix pipe: A(16x4) x ones(4x16) via
// V_WMMA_F32_16X16X4_F32. Lane L<16 supplies A[L,0]=x_L (K=1 -> 0); lane
// L>=16 supplies A[L-16,2]=x_L. Row-sum D[m,*] = x_m + x_{m+16}. Each lane
// sums its 8 D registers (rows 0..7 for lanes<16, rows 8..15 otherwise),
// then one shfl_xor(16) completes the 32-lane sum. EXEC must be all ones.
__device__ __forceinline__ float wave_sum_wmma(float x) {
  v2f a; a.x = x; a.y = 0.0f;
  v2f b; b.x = 1.0f; b.y = 1.0f;   // B = all-ones
  v8f c = {};
  c = __builtin_amdgcn_wmma_f32_16x16x4_f32(false, a, false, b,
                                            (short)0, c, false, false);
  float s = c[0] + c[1] + c[2] + c[3] + c[4] + c[5] + c[6] + c[7];
  s += __shfl_xor(s, 16, 32);
  return s;  // full 32-lane sum, replicated
}

__device__ __forceinline__ void block_partial_sum(float contrib, float* partials) {
  __shared__ float sh[8];
  float ws = wave_sum_wmma(contrib);
  int tid = threadIdx.x;
  if ((tid & 31) == 0) sh[tid >> 5] = ws;
  __syncthreads();
  if (tid == 0) {
    float s = 0.0f;
    for (int j = 0; j < 8; ++j) s += sh[j];   // fixed order -> deterministic
    partials[blockIdx.x] = s;
  }
}

__global__ void vnl_zero(unsigned* histHi, unsigned* histLo, unsigned* scalars) {
  int i = blockIdx.x * blockDim.x + threadIdx.x;
  if (i < 65536) { histHi[i] = 0u; histLo[i] = 0u; }
  if (i < 16) scalars[i] = (i == 1) ? INVALID_BITS : 0u;  // [1]=selBin sentinel
}

__global__ void __launch_bounds__(256)
vnl_compute(const float* __restrict__ gt, const float* __restrict__ pr,
            const float* __restrict__ fxp, const float* __restrict__ fyp,
            const int* __restrict__ p1x, const int* __restrict__ p1y,
            const int* __restrict__ p2x, const int* __restrict__ p2y,
            const int* __restrict__ p3x, const int* __restrict__ p3y,
            unsigned* __restrict__ lossbits, unsigned* __restrict__ histHi,
            float* __restrict__ partials, unsigned* __restrict__ scalars,
            int G, int NB) {
  int tid = threadIdx.x;
  int i = blockIdx.x * blockDim.x + tid;
  bool valid = i < NB;
  bool mask = false;
  float loss = 0.0f;

  if (valid) {
    int b = i / G;
    int g = i - b * G;
    float fx = fxp[0], fy = fyp[0];
    int px[3] = { p1x[g], p2x[g], p3x[g] };
    int py[3] = { p1y[g], p2y[g], p3y[g] };
    const float* gtb = gt + (size_t)b * (Hc * Wc);
    const float* prb = pr + (size_t)b * (Hc * Wc);

    float Gm[3][3], Pm[3][3];                 // [coord][point]
    for (int k = 0; k < 3; ++k) {
      int off = py[k] * Wc + px[k];
      float dg = gtb[off];
      float dp = prb[off];
      float u = (float)(px[k] - Wc / 2);
      float v = (float)(py[k] - Hc / 2);
      Gm[0][k] = u * fabsf(dg) / fx;  Gm[1][k] = v * fabsf(dg) / fy;  Gm[2][k] = dg;
      Pm[0][k] = u * fabsf(dp) / fx;  Pm[1][k] = v * fabsf(dp) / fy;  Pm[2][k] = dp;
    }

    // Reference quirk: zmask (pred z==0 per point k) broadcasts over the
    // COORD axis -> coordinate row k of all points becomes 1e-4.
    bool zm[3] = { Pm[2][0] == 0.0f, Pm[2][1] == 0.0f, Pm[2][2] == 0.0f };
    for (int k = 0; k < 3; ++k)
      if (zm[k]) { Pm[k][0] = 1e-4f; Pm[k][1] = 1e-4f; Pm[k][2] = 1e-4f; }

    // GT edge vectors: j=0 -> P1-P0, j=1 -> P2-P0, j=2 -> P2-P1
    float d[3][3];
    for (int c = 0; c < 3; ++c) {
      d[0][c] = Gm[c][1] - Gm[c][0];
      d[1][c] = Gm[c][2] - Gm[c][0];
      d[2][c] = Gm[c][2] - Gm[c][1];
    }
    float nrm[3];
    float dots[3][3];
    for (int a = 0; a < 3; ++a)
      for (int bb = 0; bb < 3; ++bb)
        dots[a][bb] = d[a][0]*d[bb][0] + d[a][1]*d[bb][1] + d[a][2]*d[bb][2];
    for (int a = 0; a < 3; ++a) nrm[a] = sqrtf(dots[a][a]);

    int ccos = 0;
    for (int a = 0; a < 3; ++a)
      for (int bb = 0; bb < 3; ++bb) {
        float ne = dots[a][bb] / (nrm[a] * nrm[bb] + 1e-8f);
        if (ne > DELTA_COS || ne < -DELTA_COS) ++ccos;
      }
    bool mask_cos = ccos > 3;
    bool mask_pad = (Gm[2][0] > DELTA_Zf) && (Gm[2][1] > DELTA_Zf) && (Gm[2][2] > DELTA_Zf);
    bool mx = fabsf(d[0][0]) < DELTA_DIFF || fabsf(d[1][0]) < DELTA_DIFF || fabsf(d[2][0]) < DELTA_DIFF;
    bool my = fabsf(d[0][1]) < DELTA_DIFF || fabsf(d[1][1]) < DELTA_DIFF || fabsf(d[2][1]) < DELTA_DIFF;
    bool mz = fabsf(d[0][2]) < DELTA_DIFF || fabsf(d[1][2]) < DELTA_DIFF || fabsf(d[2][2]) < DELTA_DIFF;
    mask = mask_pad && !((mx && my && mz) || mask_cos);

    if (mask) {
      // gt normal = cross(P1-P0, P2-P0)
      float gn0 = d[0][1]*d[1][2] - d[0][2]*d[1][1];
      float gn1 = d[0][2]*d[1][0] - d[0][0]*d[1][2];
      float gn2 = d[0][0]*d[1][1] - d[0][1]*d[1][0];
      // pred edges (after z-fix)
      float e0[3], e1[3];
      for (int c = 0; c < 3; ++c) {
        e0[c] = Pm[c][1] - Pm[c][0];
        e1[c] = Pm[c][2] - Pm[c][0];
      }
      float dn0 = e0[1]*e1[2] - e0[2]*e1[1];
      float dn1 = e0[2]*e1[0] - e0[0]*e1[2];
      float dn2 = e0[0]*e1[1] - e0[1]*e1[0];
      float gl = sqrtf(gn0*gn0 + gn1*gn1 + gn2*gn2);
      float dl = sqrtf(dn0*dn0 + dn1*dn1 + dn2*dn2);
      if (gl == 0.0f) gl = 0.01f;
      if (dl == 0.0f) dl = 0.01f;
      loss = fabsf(gn0/gl - dn0/dl) + fabsf(gn1/gl - dn1/dl) + fabsf(gn2/gl - dn2/dl);
    }
  }

  unsigned bits = (valid && mask) ? __float_as_uint(loss) : INVALID_BITS;
  if (valid) lossbits[i] = bits;
  if (valid && mask) atomicAdd(&histHi[bits >> 16], 1u);

  unsigned long long bal = __ballot(valid && mask);
  if ((tid & 31) == 0) atomicAdd(&scalars[0], (unsigned)__popcll(bal));

  float contrib = (valid && mask) ? loss : 0.0f;
  block_partial_sum(contrib, partials);
}

// Deterministic fixed-order reduction of per-block partials into one float.
__global__ void __launch_bounds__(256)
vnl_reduce(const float* __restrict__ partials, int n, float* __restrict__ out) {
  int tid = threadIdx.x;
  float acc = 0.0f;
  for (int j = tid; j < n; j += 256) acc += partials[j];
  __shared__ float sh[8];
  float ws = wave_sum_wmma(acc);
  if ((tid & 31) == 0) sh[tid >> 5] = ws;
  __syncthreads();
  if (tid == 0) {
    float s = 0.0f;
    for (int j = 0; j < 8; ++j) s += sh[j];
    *out = s;
  }
}

__global__ void vnl_scan_hi(const unsigned* __restrict__ histHi, unsigned* scalars) {
  if (threadIdx.x != 0 || blockIdx.x != 0) return;
  unsigned n = scalars[0];
  if (n == 0u) return;
  unsigned k = n >> 2;               // int(n*0.25) == n/4 exactly
  unsigned cum = 0;
  for (unsigned bin = 0; bin < 65536u; ++bin) {
    unsigned c = histHi[bin];
    if (cum + c > k) { scalars[1] = bin; scalars[2] = cum; return; }
    cum += c;
  }
}

__global__ void vnl_hist_lo(const unsigned* __restrict__ lossbits, int NB,
                            const unsigned* __restrict__ scalars,
                            unsigned* __restrict__ histLo) {
  int i = blockIdx.x * blockDim.x + threadIdx.x;
  if (i >= NB) return;
  unsigned bits = lossbits[i];
  if (bits == INVALID_BITS) return;
  if ((bits >> 16) == scalars[1]) atomicAdd(&histLo[bits & 0xFFFFu], 1u);
}

__global__ void vnl_scan_lo(const unsigned* __restrict__ histLo, unsigned* scalars) {
  if (threadIdx.x != 0 || blockIdx.x != 0) return;
  unsigned n = scalars[0];
  if (n == 0u) return;
  unsigned k = n >> 2;
  unsigned cum = scalars[2];
  unsigned sel = scalars[1];
  for (unsigned lo = 0; lo < 65536u; ++lo) {
    unsigned c = histLo[lo];
    if (cum + c > k) {
      scalars[3] = (sel << 16) | lo;  // T = exact k-th smallest bit pattern
      scalars[4] = cum;               // count strictly below T
      return;
    }
    cum += c;
  }
}

__global__ void __launch_bounds__(256)
vnl_sum_less(const unsigned* __restrict__ lossbits, int NB,
             const unsigned* __restrict__ scalars, float* __restrict__ partials) {
  int tid = threadIdx.x;
  int i = blockIdx.x * blockDim.x + tid;
  unsigned T = scalars[3];           // loss >= 0 so uint compare == float compare
  float contrib = 0.0f;
  if (i < NB) {
    unsigned bits = lossbits[i];
    if (bits != INVALID_BITS && bits < T) contrib = __uint_as_float(bits);
  }
  block_partial_sum(contrib, partials);
}

__global__ void vnl_final(const unsigned* __restrict__ scalars, float* __restrict__ out) {
  if (threadIdx.x != 0 || blockIdx.x != 0) return;
  unsigned n = scalars[0];
  if (n == 0u) { out[0] = 0.0f; return; }
  unsigned k = n >> 2;
  float sumAll  = __uint_as_float(scalars[8]);
  float sumLess = __uint_as_float(scalars[9]);
  unsigned cntLess = scalars[4];
  float T = __uint_as_float(scalars[3]);
  float excluded = sumLess + (float)(k - cntLess) * T;
  out[0] = (sumAll - excluded) / (float)(n - k);
}

extern "C" void kernel_launch(void* const* d_in, const int* in_sizes, int n_in,
                              void* d_out, int out_size, void* d_ws, size_t ws_size,
                              hipStream_t stream) {
  const float* gt  = (const float*)d_in[0];
  const float* pr  = (const float*)d_in[1];
  const float* fx  = (const float*)d_in[2];
  const float* fy  = (const float*)d_in[3];
  const int* p1x = (const int*)d_in[4];
  const int* p1y = (const int*)d_in[5];
  const int* p2x = (const int*)d_in[6];
  const int* p2y = (const int*)d_in[7];
  const int* p3x = (const int*)d_in[8];
  const int* p3y = (const int*)d_in[9];

  int G  = in_sizes[4];
  int NB = Bc * G;
  int nBlocks = (NB + 255) / 256;

  char* ws = (char*)d_ws;
  size_t off = 0;
  unsigned* lossbits = (unsigned*)(ws + off); off += (size_t)NB * 4; off = (off + 255) & ~(size_t)255;
  unsigned* histHi   = (unsigned*)(ws + off); off += 65536u * 4;
  unsigned* histLo   = (unsigned*)(ws + off); off += 65536u * 4;
  unsigned* scalars  = (unsigned*)(ws + off); off += 256;
  float*    partials = (float*)   (ws + off); off += (size_t)nBlocks * 4;

  vnl_zero<<<256, 256, 0, stream>>>(histHi, histLo, scalars);
  vnl_compute<<<nBlocks, 256, 0, stream>>>(gt, pr, fx, fy,
                                           p1x, p1y, p2x, p2y, p3x, p3y,
                                           lossbits, histHi, partials, scalars, G, NB);
  vnl_reduce<<<1, 256, 0, stream>>>(partials, nBlocks, (float*)&scalars[8]);
  vnl_scan_hi<<<1, 1, 0, stream>>>(histHi, scalars);
  vnl_hist_lo<<<nBlocks, 256, 0, stream>>>(lossbits, NB, scalars, histLo);
  vnl_scan_lo<<<1, 1, 0, stream>>>(histLo, scalars);
  vnl_sum_less<<<nBlocks, 256, 0, stream>>>(lossbits, NB, scalars, partials);
  vnl_reduce<<<1, 256, 0, stream>>>(partials, nBlocks, (float*)&scalars[9]);
  vnl_final<<<1, 1, 0, stream>>>(scalars, (float*)d_out);
}